// ConvSBS_11450382812148
// MI455X (gfx1250) — compile-verified
//
#include <hip/hip_runtime.h>

// CDNA5 / gfx1250 fused ConvSBS (tensor-network string-bond conv, 2x2 ring).
//
// Stage 1 (per source pixel, per core):  t[qlr] = sum_k X[k] * W[qlr][k]
//   with X[k=i*4+j] = ch0[i]*ch1[j]  ->  WMMA f32 16x16x4, M=16 pixels,
//   N=16 qlr columns (8 tiles), K=16 (4 chained x4 steps).
// Stage 2 (per output pixel, 16 channels): out = trace(A_p B_q C_r D_s)
//   over a 2x2 stencil of stage-1 results held in LDS.

typedef __attribute__((ext_vector_type(2))) float v2f;
typedef __attribute__((ext_vector_type(4))) float v4f;
typedef __attribute__((ext_vector_type(8))) float v8f;

#define BATCH 8
#define H 128
#define W 128
#define OH 127
#define OW 127
#define TILE 8           // 8x8 output pixels per workgroup
#define SRC 9            // 9x9 source halo -> 81 pixels
#define NSLOT 96         // 81 padded to 6 groups of 16 (WMMA M)
#define PIXSTRIDE 516    // 4 cores * 128 floats + 4 pad (bank-conflict break)

__global__ __launch_bounds__(1024, 1)
void convsbs_fused_kernel(const float* __restrict__ channels,
                          const float* __restrict__ cores,
                          float* __restrict__ out) {
  __shared__ float inter[NSLOT * PIXSTRIDE];   // ~194 KB of the 320 KB WGP LDS

  const int b   = blockIdx.z;
  const int ty0 = blockIdx.y * TILE;
  const int tx0 = blockIdx.x * TILE;

  const float* __restrict__ ch0 = channels;
  const float* __restrict__ ch1 = channels + (long)BATCH * H * W * 4;

  const int tid  = (int)threadIdx.x;
  const int wv   = tid >> 5;      // wave id 0..31 (wave32)
  const int lane = tid & 31;

  // ---------------- Stage 1: WMMA f32 16x16x4 per-pixel GEMM ----------------
  // 24 active waves: wave -> (pixel group g = wv>>2 in 0..5, core = wv&3).
  if (wv < 24) {
    const int g    = wv >> 2;
    const int core = wv & 3;
    const int m    = lane & 15;   // WMMA M row (pixel within group)
    const int half = lane >> 4;   // K-half select (A/B fragment layout)

    // Source pixel slot; slots 81..95 are padding (computed but never read).
    int s = g * 16 + m;
    int sc = s > 80 ? 80 : s;                  // clamp for safe global loads
    int sy = sc / SRC, sx = sc % SRC;
    int gy = ty0 + sy; gy = gy > (H - 1) ? (H - 1) : gy;
    int gx = tx0 + sx; gx = gx > (W - 1) ? (W - 1) : gx;
    const long pixoff = ((((long)b * H + gy) * W) + gx) * 4;

    // A fragment inputs: X[m][kt*4 + 2*half + {0,1}] = ch0[m][kt]*ch1[m][2*half+{0,1}]
    const v4f c0 = *(const v4f*)(ch0 + pixoff);
    const v2f c1 = *(const v2f*)(ch1 + pixoff + 2 * half);

    v2f afrag[4];
#pragma unroll
    for (int kt = 0; kt < 4; ++kt) {
      afrag[kt][0] = c0[kt] * c1[0];
      afrag[kt][1] = c0[kt] * c1[1];
    }

    // Weight base for this core: W[col][k] = cores[core][col*16 + k], col=(q*8+l)*8+r
    const float* __restrict__ wbase = cores + core * 2048;
    const int sbase = g * 16 + 8 * half;       // D-fragment row base in LDS

#pragma unroll
    for (int nt = 0; nt < 8; ++nt) {           // 8 N-tiles cover 128 qlr columns
      v8f acc = {};
#pragma unroll
      for (int kt = 0; kt < 4; ++kt) {
        // B fragment: B[k][n] = W[nt*16+n][kt*4 + 2*half + {0,1}] (contiguous pair)
        const v2f bfrag =
            *(const v2f*)(wbase + (nt * 16 + m) * 16 + kt * 4 + 2 * half);
        acc = __builtin_amdgcn_wmma_f32_16x16x4_f32(
            false, afrag[kt], false, bfrag, (short)0, acc, false, false);
      }
      // D layout: lane -> col = nt*16 + (lane&15); VGPR v -> pixel row v + 8*half.
      const int colbase = core * 128 + nt * 16 + m;
#pragma unroll
      for (int v = 0; v < 8; ++v)
        inter[(sbase + v) * PIXSTRIDE + colbase] = acc[v];
    }
  }

  __syncthreads();

  // ---------------- Stage 2: ring trace over 2x2 stencil --------------------
  // thread -> (output pixel in 8x8 tile, output channel pqrs)
  const int pix = tid >> 4;
  const int chn = tid & 15;
  const int oyl = pix >> 3, oxl = pix & 7;
  const int p  = (chn >> 3) & 1;
  const int q  = (chn >> 2) & 1;
  const int r  = (chn >> 1) & 1;
  const int s2 = chn & 1;

  const int sA = oyl * SRC + oxl;              // core0 @ (oy, ox)
  const float* __restrict__ Am = &inter[(sA)*PIXSTRIDE + 0 * 128 + p * 64];
  const float* __restrict__ Bm = &inter[(sA + 1) * PIXSTRIDE + 1 * 128 + q * 64];
  const float* __restrict__ Cm = &inter[(sA + SRC) * PIXSTRIDE + 2 * 128 + r * 64];
  const float* __restrict__ Dm = &inter[(sA + SRC + 1) * PIXSTRIDE + 3 * 128 + s2 * 64];

  // acc = sum_{i,l} ((A*B)*C)[i,l] * D[l,i], streamed one E-row at a time.
  float acc = 0.f;
  for (int i = 0; i < 8; ++i) {
    float e[8];
#pragma unroll
    for (int k = 0; k < 8; ++k) e[k] = 0.f;
#pragma unroll
    for (int j = 0; j < 8; ++j) {
      const float av = Am[i * 8 + j];
#pragma unroll
      for (int k = 0; k < 8; ++k) e[k] = fmaf(av, Bm[j * 8 + k], e[k]);
    }
#pragma unroll
    for (int l = 0; l < 8; ++l) {
      float gv = 0.f;
#pragma unroll
      for (int k = 0; k < 8; ++k) gv = fmaf(e[k], Cm[k * 8 + l], gv);
      acc = fmaf(gv, Dm[l * 8 + i], acc);
    }
  }

  const int oy = ty0 + oyl;
  const int ox = tx0 + oxl;
  if (oy < OH && ox < OW)
    out[((((long)b * OH + oy) * OW) + ox) * 16 + chn] = acc;
}

extern "C" void kernel_launch(void* const* d_in, const int* in_sizes, int n_in,
                              void* d_out, int out_size, void* d_ws, size_t ws_size,
                              hipStream_t stream) {
  (void)in_sizes; (void)n_in; (void)d_ws; (void)ws_size; (void)out_size;
  const float* channels = (const float*)d_in[0];  // [2,8,128,128,4] f32
  const float* cores    = (const float*)d_in[1];  // [4,2,8,8,4,4]   f32
  float* out            = (float*)d_out;          // [8,127,127,16]  f32

  dim3 grid((OW + TILE - 1) / TILE, (OH + TILE - 1) / TILE, BATCH); // 16x16x8
  dim3 block(1024);
  hipLaunchKernelGGL(convsbs_fused_kernel, grid, block, 0, stream,
                     channels, cores, out);
}